// ProxyNCA_prob_mixup_72748156059751
// MI455X (gfx1250) — compile-verified
//
#include <hip/hip_runtime.h>
#include <math.h>

typedef __attribute__((ext_vector_type(2))) float v2f;
typedef __attribute__((ext_vector_type(8))) float v8f;

#define N_PTS   256
#define N_CLS   100
#define DIMS    512
#define LDIP    112                      // N_CLS padded to multiple of 16
#define SCALE_F 3.0f
#define EPS_N   1e-12f
#define M_PAIRS (N_PTS*(N_PTS-1)/2)      // 32640
#define N_JOBS  (N_PTS + M_PAIRS)        // 32896
#define N_PAD   (LDIP - N_CLS)           // 12 zero rows appended to Pn

__device__ __forceinline__ int triu_off(int i) {  // # pairs with first index < i
    return i * (2*N_PTS - 1 - i) / 2;
}

// ---------------------------------------------------------------------------
// 1) Row L2-normalize + scale by 3.
//    Blocks 0..99    : proxies -> Pn rows 0..99
//    Blocks 100..355 : X       -> Xn rows 0..255
//    Blocks 356..367 : write zero pad rows Pn[100..111] (so GEMM needs no mask)
// ---------------------------------------------------------------------------
__global__ void normalize_kernel(const float* __restrict__ X,
                                 const float* __restrict__ P,
                                 float* __restrict__ Xn,
                                 float* __restrict__ Pn) {
    int row = blockIdx.x;
    int t = threadIdx.x;                 // 256 threads = 8 waves

    if (row >= N_CLS + N_PTS) {          // zero-pad rows of Pn
        float* dst = Pn + (size_t)(N_CLS + (row - N_CLS - N_PTS)) * DIMS;
        for (int k = t; k < DIMS; k += 256) dst[k] = 0.0f;
        return;
    }

    __shared__ float red[9];
    const float* src;
    float* dst;
    if (row < N_CLS) { src = P + (size_t)row * DIMS;           dst = Pn + (size_t)row * DIMS; }
    else             { src = X + (size_t)(row - N_CLS) * DIMS; dst = Xn + (size_t)(row - N_CLS) * DIMS; }

    float s = 0.f;
    for (int k = t; k < DIMS; k += 256) { float v = src[k]; s += v * v; }
    for (int off = 16; off >= 1; off >>= 1) s += __shfl_xor(s, off, 32);
    int wave = t >> 5, lane = t & 31;
    if (lane == 0) red[wave] = s;
    __syncthreads();
    if (t == 0) {
        float tot = 0.f;
        for (int w = 0; w < 8; w++) tot += red[w];
        red[8] = SCALE_F / fmaxf(sqrtf(tot), EPS_N);
    }
    __syncthreads();
    float sc = red[8];
    for (int k = t; k < DIMS; k += 256) dst[k] = src[k] * sc;
}

// ---------------------------------------------------------------------------
// 2) Out[m,n] = sum_k A[m,k]*B[n,k]  via V_WMMA_F32_16X16X4_F32.
//    One wave per 16x16 tile; B already padded so no masking in the K-loop:
//    inner loop = 2x global_load_b64 + 1x v_wmma.
//    A layout (16x4): lane l(0-15) holds A[m=l][K=0,1]; lane l+16 holds K=2,3.
//    B layout (4x16): lane l(0-15) holds B[n=l][K=0,1]; lane l+16 holds K=2,3.
//    C/D layout: VGPR i -> M = i + 8*(lane>=16), N = lane&15.
// ---------------------------------------------------------------------------
__global__ void gemm_nt_wmma(const float* __restrict__ A,
                             const float* __restrict__ B,
                             float* __restrict__ Out,
                             int ldOut) {
    int lane = threadIdx.x;              // blockDim.x == 32
    int m0 = blockIdx.x * 16;
    int n0 = blockIdx.y * 16;
    int r    = lane & 15;
    int koff = (lane >> 4) * 2;          // 0 or 2

    const float* aRow = A + (size_t)(m0 + r) * DIMS + koff;
    const float* bRow = B + (size_t)(n0 + r) * DIMS + koff;

    v8f acc = {};
    for (int k = 0; k < DIMS; k += 4) {
        v2f a = *(const v2f*)(aRow + k);
        v2f b = *(const v2f*)(bRow + k);
        acc = __builtin_amdgcn_wmma_f32_16x16x4_f32(
            /*neg_a=*/false, a, /*neg_b=*/false, b,
            /*c_mod=*/(short)0, acc, /*reuse_a=*/false, /*reuse_b=*/false);
    }

    int rowBase = m0 + ((lane >> 4) << 3);
    int col     = n0 + r;
    #pragma unroll
    for (int i = 0; i < 8; i++)
        Out[(size_t)(rowBase + i) * ldOut + col] = acc[i];
}

// ---------------------------------------------------------------------------
// 3) One wave per job. job < N_PTS: real-row loss. Else: pair loss via the
//    Gram-matrix factorization (no virtual vectors ever materialized).
// ---------------------------------------------------------------------------
__global__ void job_kernel(const float* __restrict__ IP,
                           const float* __restrict__ G,
                           const int*   __restrict__ T,
                           float* __restrict__ lossArr,
                           float* __restrict__ cntArr) {
    int job  = blockIdx.x * (blockDim.x >> 5) + (threadIdx.x >> 5);
    int lane = threadIdx.x & 31;
    if (job >= N_JOBS) return;

    float loss = 0.f, cnt = 0.f;

    if (job < N_PTS) {
        const float* row = IP + job * LDIP;
        float s0[4], mx = -1e30f;
        #pragma unroll
        for (int j = 0; j < 4; j++) {
            int c = lane + 32 * j;
            float s = -1e30f;
            if (c < N_CLS) s = -fmaxf(18.0f - 2.0f * row[c], 0.0f);
            s0[j] = s; mx = fmaxf(mx, s);
        }
        for (int off = 16; off >= 1; off >>= 1) mx = fmaxf(mx, __shfl_xor(mx, off, 32));
        float sum = 0.f;
        #pragma unroll
        for (int j = 0; j < 4; j++) {
            int c = lane + 32 * j;
            if (c < N_CLS) sum += expf(s0[j] - mx);
        }
        for (int off = 16; off >= 1; off >>= 1) sum += __shfl_xor(sum, off, 32);
        float lse = mx + logf(sum);
        float st  = -fmaxf(18.0f - 2.0f * row[T[job]], 0.0f);
        loss = lse - st;
        cnt  = 1.0f;
    } else {
        int m = job - N_PTS;
        double b = 2.0 * N_PTS - 1.0;
        int i1 = (int)((b - sqrt(b * b - 8.0 * (double)m)) * 0.5);
        while (triu_off(i1 + 1) <= m) i1++;
        while (i1 > 0 && triu_off(i1) > m) i1--;
        int i2 = i1 + 1 + (m - triu_off(i1));

        int c1 = T[i1], c2 = T[i2];
        if (c1 != c2) {
            const float* r1 = IP + i1 * LDIP;
            const float* r2 = IP + i2 * LDIP;
            float X1P1 = fminf(fmaxf(r1[c1], -1.f), 1.f);
            float X1P2 = fminf(fmaxf(r1[c2], -1.f), 1.f);
            float X2P1 = fminf(fmaxf(r2[c1], -1.f), 1.f);
            float X2P2 = fminf(fmaxf(r2[c2], -1.f), 1.f);
            float lam = (X2P2 - X2P1) / (X2P2 - X2P1 + X1P1 - X1P2);
            lam = fminf(fmaxf(lam, 0.3f), 0.7f);
            float om = 1.0f - lam;

            // ||lam*Xn1 + om*Xn2||^2 = 9(lam^2+om^2) + 2 lam om G[i1,i2]
            float g   = G[i1 * N_PTS + i2];
            float nrm = sqrtf(9.0f * (lam * lam + om * om) + 2.0f * lam * om * g);
            float inv = SCALE_F / fmaxf(nrm, EPS_N);

            float s0[4], mx = -1e30f;
            #pragma unroll
            for (int j = 0; j < 4; j++) {
                int c = lane + 32 * j;
                float s = -1e30f;
                if (c < N_CLS) {
                    float v = inv * (lam * r1[c] + om * r2[c]);
                    s = -fmaxf(18.0f - 2.0f * v, 0.0f);
                }
                s0[j] = s; mx = fmaxf(mx, s);
            }
            for (int off = 16; off >= 1; off >>= 1) mx = fmaxf(mx, __shfl_xor(mx, off, 32));
            float sum = 0.f;
            #pragma unroll
            for (int j = 0; j < 4; j++) {
                int c = lane + 32 * j;
                if (c < N_CLS) sum += expf(s0[j] - mx);
            }
            for (int off = 16; off >= 1; off >>= 1) sum += __shfl_xor(sum, off, 32);
            float lse = mx + logf(sum);

            float vA = inv * (lam * r1[c1] + om * r2[c1]);
            float vB = inv * (lam * r1[c2] + om * r2[c2]);
            float sA = -fmaxf(18.0f - 2.0f * vA, 0.0f);
            float sB = -fmaxf(18.0f - 2.0f * vB, 0.0f);
            loss = lse - lam * sA - om * sB;
            cnt  = 1.0f;
        }
    }
    if (lane == 0) { lossArr[job] = loss; cntArr[job] = cnt; }
}

// ---------------------------------------------------------------------------
// 4) Deterministic fixed-order reduction: out = sum(loss) / sum(cnt).
// ---------------------------------------------------------------------------
__global__ void reduce_kernel(const float* __restrict__ lossArr,
                              const float* __restrict__ cntArr,
                              float* __restrict__ out) {
    __shared__ float sL[256], sC[256];
    int t = threadIdx.x;                 // 256 threads
    float l = 0.f, c = 0.f;
    for (int i = t; i < N_JOBS; i += 256) { l += lossArr[i]; c += cntArr[i]; }
    sL[t] = l; sC[t] = c;
    __syncthreads();
    for (int s = 128; s > 0; s >>= 1) {
        if (t < s) { sL[t] += sL[t + s]; sC[t] += sC[t + s]; }
        __syncthreads();
    }
    if (t == 0) out[0] = sL[0] / sC[0];
}

// ---------------------------------------------------------------------------
extern "C" void kernel_launch(void* const* d_in, const int* in_sizes, int n_in,
                              void* d_out, int out_size, void* d_ws, size_t ws_size,
                              hipStream_t stream) {
    const float* X       = (const float*)d_in[0];   // 256*512
    const float* proxies = (const float*)d_in[1];   // 100*512
    const int*   T       = (const int*)d_in[2];     // 256
    (void)in_sizes; (void)n_in; (void)out_size; (void)ws_size;

    float* ws      = (float*)d_ws;
    float* Pn      = ws;                            // 112*512 (12 zero pad rows)
    float* Xn      = Pn + (size_t)LDIP  * DIMS;     // 256*512
    float* IP      = Xn + (size_t)N_PTS * DIMS;     // 256*112
    float* G       = IP + (size_t)N_PTS * LDIP;     // 256*256
    float* lossArr = G  + (size_t)N_PTS * N_PTS;    // 32896
    float* cntArr  = lossArr + N_JOBS;              // 32896

    // 1) normalize (356 data rows + 12 pad rows of Pn)
    normalize_kernel<<<N_CLS + N_PTS + N_PAD, 256, 0, stream>>>(X, proxies, Xn, Pn);

    // 2) IP = Xn @ Pn^T  (256 x 112; cols 100..111 are exact zeros via pad rows)
    gemm_nt_wmma<<<dim3(N_PTS / 16, LDIP / 16), 32, 0, stream>>>(Xn, Pn, IP, LDIP);

    // 3) G = Xn @ Xn^T  (256 x 256)
    gemm_nt_wmma<<<dim3(N_PTS / 16, N_PTS / 16), 32, 0, stream>>>(Xn, Xn, G, N_PTS);

    // 4) per-job losses (one wave32 per job, 8 waves per block)
    job_kernel<<<(N_JOBS + 7) / 8, 256, 0, stream>>>(IP, G, T, lossArr, cntArr);

    // 5) final deterministic reduction
    reduce_kernel<<<1, 256, 0, stream>>>(lossArr, cntArr, (float*)d_out);
}